// GCNRecommender_33131377721642
// MI455X (gfx1250) — compile-verified
//
#include <hip/hip_runtime.h>

#define DIM 32

typedef float v2f __attribute__((ext_vector_type(2)));
typedef float v8f __attribute__((ext_vector_type(8)));

// ---------------------------------------------------------------------------
// Zero an f32 buffer (graph-capture safe; no hipMemsetAsync needed)
// ---------------------------------------------------------------------------
__global__ __launch_bounds__(256) void zero_f32(float* __restrict__ p, int n) {
    int i = blockIdx.x * blockDim.x + threadIdx.x;
    if (i < n) p[i] = 0.0f;
}

// ---------------------------------------------------------------------------
// SpMM: y[row] += val * x[col]  (one edge per wave32, one feature per lane)
// Index/value loads are wave-uniform; gather + atomic scatter are coalesced
// 128B transactions. x (12.8 MB) lives in the 192 MB L2, so this runs at
// L2 bandwidth, not HBM-gather speed.
// ---------------------------------------------------------------------------
__global__ __launch_bounds__(256) void spmm_scatter(
        const int*   __restrict__ rows,
        const int*   __restrict__ cols,
        const float* __restrict__ vals,
        const float* __restrict__ x,
        float*       __restrict__ y,
        int num_edges) {
    const int lane   = threadIdx.x & 31;
    const int total  = num_edges << 5;               // edges * 32 lanes
    const int stride = gridDim.x * blockDim.x;       // multiple of 32
    for (int idx = blockIdx.x * blockDim.x + threadIdx.x; idx < total; idx += stride) {
        const int   e = idx >> 5;                    // same edge for all 32 lanes
        const int   r = rows[e];
        const int   c = cols[e];
        const float v = vals[e];
        const float m = v * x[c * DIM + lane];
        unsafeAtomicAdd(&y[r * DIM + lane], m);      // global_atomic_add_f32
    }
}

// ---------------------------------------------------------------------------
// Dense layer: out = relu(x @ W^T + b), all f32, on the WMMA pipe.
// One wave computes a 16x32 output tile with 2 column tiles x 8 chained
// V_WMMA_F32_16X16X4_F32 (K=4 each -> K=32).
//
// Fragment layouts per CDNA5 ISA 7.12.2:
//   A (16x4 f32, 2 VGPRs): lanes 0-15 hold K={0,1}, lanes 16-31 hold K={2,3}
//   B (4x16 f32, 2 VGPRs): same K split, N = lane&15; B[k][n] = W[n][k]
//   C/D (16x16 f32, 8 VGPRs): VGPR j -> row j (lanes 0-15) / row j+8 (16-31)
// ---------------------------------------------------------------------------
__global__ __launch_bounds__(256) void dense_relu_wmma(
        const float* __restrict__ x,
        const float* __restrict__ w,      // [32,32] row-major, out = x @ w^T
        const float* __restrict__ bias,   // [32]
        float*       __restrict__ out,
        int num_tiles) {
    const int wave = (blockIdx.x * blockDim.x + threadIdx.x) >> 5;
    const int lane = threadIdx.x & 31;
    if (wave >= num_tiles) return;        // wave-uniform: EXEC stays all-1s

    const int row0    = wave * 16;
    const int halfsel = lane >> 4;        // 0: lanes 0-15, 1: lanes 16-31
    const int l15     = lane & 15;

    // A fragments for all 8 K-chunks: lane reads its row's K = 4*kb + 2*halfsel {+0,+1}
    v2f a[8];
    const float* xrow = x + (row0 + l15) * DIM;
    #pragma unroll
    for (int kb = 0; kb < 8; ++kb) {
        const int k0 = kb * 4 + halfsel * 2;
        a[kb][0] = xrow[k0];
        a[kb][1] = xrow[k0 + 1];
    }

    #pragma unroll
    for (int nt = 0; nt < 2; ++nt) {
        const int   n  = nt * 16 + l15;       // output column owned by this lane
        const float bn = bias[n];
        v8f c = {bn, bn, bn, bn, bn, bn, bn, bn};   // bias pre-loaded into C

        const float* wrow = w + n * DIM;      // B[k][n] = W[n][k]
        #pragma unroll
        for (int kb = 0; kb < 8; ++kb) {
            const int k0 = kb * 4 + halfsel * 2;
            v2f bfrag;
            bfrag[0] = wrow[k0];
            bfrag[1] = wrow[k0 + 1];
            // (neg_a, A, neg_b, B, c_mod, C, reuse_a, reuse_b)
            c = __builtin_amdgcn_wmma_f32_16x16x4_f32(
                    false, a[kb], false, bfrag, (short)0, c, false, false);
        }

        // ReLU + store: VGPR j holds row (j + 8*halfsel), column n
        #pragma unroll
        for (int j = 0; j < 8; ++j) {
            const int   m = j + halfsel * 8;
            const float v = c[j] > 0.0f ? c[j] : 0.0f;
            out[(row0 + m) * DIM + n] = v;
        }
    }
}

// ---------------------------------------------------------------------------
// Host launcher: layer = zero -> spmm -> dense(relu), twice.
// Workspace: two ping-pong [N,32] f32 buffers (25.6 MB).
// ---------------------------------------------------------------------------
extern "C" void kernel_launch(void* const* d_in, const int* in_sizes, int n_in,
                              void* d_out, int out_size, void* d_ws, size_t ws_size,
                              hipStream_t stream) {
    const int*   rows = (const int*)  d_in[0];
    const int*   cols = (const int*)  d_in[1];
    const float* vals = (const float*)d_in[2];
    const float* emb  = (const float*)d_in[3];
    const float* w0   = (const float*)d_in[4];
    const float* b0   = (const float*)d_in[5];
    const float* w1   = (const float*)d_in[6];
    const float* b1   = (const float*)d_in[7];

    const int E = in_sizes[0];
    const int N = in_sizes[3] / DIM;

    float* bufA = (float*)d_ws;                 // SpMM accumulator
    float* bufB = bufA + (size_t)N * DIM;       // dense layer-1 output

    const int nElems = N * DIM;
    const dim3 zgrid((nElems + 255) / 256);

    int sblocks = (E * DIM + 255) / 256;        // one edge per wave, grid-stride
    if (sblocks > 8192) sblocks = 8192;

    const int  tiles = (N + 15) / 16;           // 16-row WMMA tiles (N%16==0 here)
    const dim3 dgrid((tiles + 7) / 8);          // 8 waves / 256-thread block

    // ---- layer 1 ----
    zero_f32      <<<zgrid,   256, 0, stream>>>(bufA, nElems);
    spmm_scatter  <<<sblocks, 256, 0, stream>>>(rows, cols, vals, emb, bufA, E);
    dense_relu_wmma<<<dgrid,  256, 0, stream>>>(bufA, w0, b0, bufB, tiles);

    // ---- layer 2 ----
    zero_f32      <<<zgrid,   256, 0, stream>>>(bufA, nElems);
    spmm_scatter  <<<sblocks, 256, 0, stream>>>(rows, cols, vals, bufB, bufA, E);
    dense_relu_wmma<<<dgrid,  256, 0, stream>>>(bufA, w1, b1, (float*)d_out, tiles);
}